// MemN2N_23519240912991
// MI455X (gfx1250) — compile-verified
//
#include <hip/hip_runtime.h>
#include <hip/hip_bf16.h>
#include <math.h>

// Problem constants (match reference)
#define NB   128      // BATCH
#define NM   200      // MEM_SIZE
#define NS   50       // SENT_SIZE
#define ND   128      // EMBED_DIM
#define NV   50000    // NUM_VOCAB
#define NHOP 3

typedef __attribute__((ext_vector_type(2))) float v2f;
typedef __attribute__((ext_vector_type(8))) float v8f;

// Position encoding, analytic. enc is built as [D,S], enc[:, -1]=1, then transposed
// to [S,D] => after transpose: enc[S-1, d] == 1 for all d.
__device__ __forceinline__ float pos_enc(int s, float di /* (d+1)-64.5 */) {
    if (s == NS - 1) return 1.0f;
    const float sj = (float)(s + 1) - 0.5f * (float)(NS + 1);   // (s+1)-25.5
    return 1.0f + (4.0f / (float)(ND * NS)) * di * sj;
}

// ---------------------------------------------------------------------------
// Kernel 1: m_all[k][b][m][d] = sum_s C[k][story[b,m,s]][d] * enc(s,d), k=0..3
// One block per (b,m); 128 threads = one per d. All 4 tables in one pass.
// ---------------------------------------------------------------------------
__global__ __launch_bounds__(128)
void embed_story_kernel(const int* __restrict__ story,
                        const float* __restrict__ C,
                        float* __restrict__ m_all) {
    __shared__ int sidx[NS];
    const int bm = blockIdx.x;            // 0 .. B*M-1
    const int d  = threadIdx.x;           // 0 .. 127
    if (d < NS) sidx[d] = story[(size_t)bm * NS + d];
    __syncthreads();

    const float di = (float)(d + 1) - 0.5f * (float)(ND + 1);
    const float* C0 = C;
    const float* C1 = C + (size_t)1 * NV * ND;
    const float* C2 = C + (size_t)2 * NV * ND;
    const float* C3 = C + (size_t)3 * NV * ND;

    float a0 = 0.f, a1 = 0.f, a2 = 0.f, a3 = 0.f;
    #pragma unroll 5
    for (int s = 0; s < NS; ++s) {
        const float  e    = pos_enc(s, di);
        const size_t base = (size_t)sidx[s] * ND + d;
        a0 = fmaf(C0[base], e, a0);
        a1 = fmaf(C1[base], e, a1);
        a2 = fmaf(C2[base], e, a2);
        a3 = fmaf(C3[base], e, a3);
    }
    const size_t stride = (size_t)NB * NM * ND;
    const size_t off    = (size_t)bm * ND + d;
    m_all[0 * stride + off] = a0;
    m_all[1 * stride + off] = a1;
    m_all[2 * stride + off] = a2;
    m_all[3 * stride + off] = a3;
}

// ---------------------------------------------------------------------------
// Kernel 2: u0[b][d] = sum_s C[0][query[b,s]][d] * enc(s,d)
// ---------------------------------------------------------------------------
__global__ __launch_bounds__(128)
void embed_query_kernel(const int* __restrict__ query,
                        const float* __restrict__ C,
                        float* __restrict__ u) {
    __shared__ int sidx[NS];
    const int b = blockIdx.x;
    const int d = threadIdx.x;
    if (d < NS) sidx[d] = query[(size_t)b * NS + d];
    __syncthreads();

    const float di = (float)(d + 1) - 0.5f * (float)(ND + 1);
    float acc = 0.f;
    #pragma unroll 5
    for (int s = 0; s < NS; ++s)
        acc = fmaf(C[(size_t)sidx[s] * ND + d], pos_enc(s, di), acc);
    u[(size_t)b * ND + d] = acc;
}

// ---------------------------------------------------------------------------
// Kernel 3 (per hop): scores -> softmax -> o_k -> gated update of u
// One block per batch row b; 256 threads = 8 wave32's.
// ---------------------------------------------------------------------------
__global__ __launch_bounds__(256)
void hop_kernel(const float* __restrict__ m_all,
                float* __restrict__ u,
                const float* __restrict__ Tw,
                const float* __restrict__ Tb,
                int hop) {
    __shared__ float u_s[ND];
    __shared__ float scores[NM];
    __shared__ float red[256];

    const int b    = blockIdx.x;
    const int tid  = threadIdx.x;
    const int wave = tid >> 5;
    const int lane = tid & 31;

    const size_t tstride = (size_t)NB * NM * ND;
    const float* m_A = m_all + (size_t)hop       * tstride + (size_t)b * NM * ND;
    const float* m_C = m_all + (size_t)(hop + 1) * tstride + (size_t)b * NM * ND;

    if (tid < ND) u_s[tid] = u[(size_t)b * ND + tid];
    __syncthreads();

    // scores[m] = dot(m_A[m,:], u)  -- one wave32 per m, 4 elems/lane
    for (int m = wave; m < NM; m += 8) {
        const float* row = m_A + (size_t)m * ND;
        float p = row[lane]      * u_s[lane]
                + row[lane + 32] * u_s[lane + 32]
                + row[lane + 64] * u_s[lane + 64]
                + row[lane + 96] * u_s[lane + 96];
        #pragma unroll
        for (int off = 16; off > 0; off >>= 1)
            p += __shfl_down(p, off, 32);
        if (lane == 0) scores[m] = p;
    }
    __syncthreads();

    // softmax over M=200 in LDS
    float sv = (tid < NM) ? scores[tid] : -3.402823e38f;
    red[tid] = sv;
    __syncthreads();
    for (int off = 128; off > 0; off >>= 1) {
        if (tid < off) red[tid] = fmaxf(red[tid], red[tid + off]);
        __syncthreads();
    }
    const float mx = red[0];
    __syncthreads();
    float ev = 0.f;
    if (tid < NM) { ev = expf(sv - mx); scores[tid] = ev; }
    red[tid] = ev;
    __syncthreads();
    for (int off = 128; off > 0; off >>= 1) {
        if (tid < off) red[tid] += red[tid + off];
        __syncthreads();
    }
    const float inv = 1.0f / red[0];
    __syncthreads();
    if (tid < NM) scores[tid] *= inv;
    __syncthreads();

    // o_k, t_k, gated update (threads 0..127, one per d)
    if (tid < ND) {
        const int d = tid;
        float o = 0.f;
        for (int m = 0; m < NM; ++m)
            o = fmaf(scores[m], m_C[(size_t)m * ND + d], o);

        float t = Tb[d];
        const float* twr = Tw + (size_t)d * ND;     // (u @ Tw^T)[d] = sum_j u[j]*Tw[d][j]
        for (int j = 0; j < ND; ++j)
            t = fmaf(twr[j], u_s[j], t);
        t = 1.0f / (1.0f + expf(-t));

        u[(size_t)b * ND + d] = (1.0f - t) * u_s[d] + o * t;
    }
}

// ---------------------------------------------------------------------------
// Kernel 4: a_hat[b,v] = sum_d u[b,d] * C3[v,d]   (M=128, N=50000, K=128)
// 1 wave per block; each block computes a 16(b) x 80(v) tile via 5 WMMA accs.
// Uses V_WMMA_F32_16X16X4_F32; u tile staged in LDS.
// ---------------------------------------------------------------------------
__global__ __launch_bounds__(32)
void ahat_wmma_kernel(const float* __restrict__ u,
                      const float* __restrict__ C3,
                      float* __restrict__ out) {
    __shared__ float uTile[16 * ND];   // 8 KB

    const int lane  = threadIdx.x;           // 0..31
    const int bTile = blockIdx.y * 16;       // row (batch) tile
    const int v0    = blockIdx.x * 80;       // col (vocab) tile: 5 x 16

    // stage the 16x128 u tile into LDS
    for (int i = lane; i < 16 * ND; i += 32)
        uTile[i] = u[(size_t)(bTile + (i >> 7)) * ND + (i & (ND - 1))];
    __syncthreads();

    const int n     = lane & 15;             // N index within 16-wide tile
    const int khalf = (lane >> 4) << 1;      // lanes 0-15 -> K{0,1}; 16-31 -> K{2,3}

    // warm the L2/L0 path for the 80 C3 rows this wave will stream
    #pragma unroll
    for (int t = 0; t < 5; ++t)
        __builtin_prefetch(C3 + (size_t)(v0 + t * 16 + n) * ND, 0, 3);

    v8f acc0 = {}, acc1 = {}, acc2 = {}, acc3 = {}, acc4 = {};

    for (int kb = 0; kb < ND; kb += 4) {
        const int kk = kb + khalf;
        // A fragment: A[m,k], m = lane&15, this lane's K pair = {kk, kk+1}
        v2f a;
        a.x = uTile[n * ND + kk];
        a.y = uTile[n * ND + kk + 1];
        // B fragments: B[k, ncol] = C3[v0 + tile*16 + ncol][k]
        const float* bp = C3 + (size_t)(v0 + n) * ND + kk;
        v2f b0, b1, b2, b3, b4;
        b0.x = bp[0 * 16 * ND]; b0.y = bp[0 * 16 * ND + 1];
        b1.x = bp[1 * 16 * ND]; b1.y = bp[1 * 16 * ND + 1];
        b2.x = bp[2 * 16 * ND]; b2.y = bp[2 * 16 * ND + 1];
        b3.x = bp[3 * 16 * ND]; b3.y = bp[3 * 16 * ND + 1];
        b4.x = bp[4 * 16 * ND]; b4.y = bp[4 * 16 * ND + 1];

        acc0 = __builtin_amdgcn_wmma_f32_16x16x4_f32(false, a, false, b0, (short)0, acc0, false, false);
        acc1 = __builtin_amdgcn_wmma_f32_16x16x4_f32(false, a, false, b1, (short)0, acc1, false, false);
        acc2 = __builtin_amdgcn_wmma_f32_16x16x4_f32(false, a, false, b2, (short)0, acc2, false, false);
        acc3 = __builtin_amdgcn_wmma_f32_16x16x4_f32(false, a, false, b3, (short)0, acc3, false, false);
        acc4 = __builtin_amdgcn_wmma_f32_16x16x4_f32(false, a, false, b4, (short)0, acc4, false, false);
    }

    // C/D layout: vgpr r -> M = r (lanes 0-15) or r+8 (lanes 16-31); N = lane&15
    const int mbase = bTile + ((lane >> 4) << 3);
    #pragma unroll
    for (int r = 0; r < 8; ++r) {
        const size_t row = (size_t)(mbase + r) * NV;
        out[row + v0 +  0 + n] = acc0[r];
        out[row + v0 + 16 + n] = acc1[r];
        out[row + v0 + 32 + n] = acc2[r];
        out[row + v0 + 48 + n] = acc3[r];
        out[row + v0 + 64 + n] = acc4[r];
    }
}

// ---------------------------------------------------------------------------
extern "C" void kernel_launch(void* const* d_in, const int* in_sizes, int n_in,
                              void* d_out, int out_size, void* d_ws, size_t ws_size,
                              hipStream_t stream) {
    const int*   story = (const int*)  d_in[0];   // [B, M, S]
    const int*   query = (const int*)  d_in[1];   // [B, S]
    const float* C     = (const float*)d_in[2];   // [4, V, D]
    const float* Tw    = (const float*)d_in[3];   // [D, D]
    const float* Tb    = (const float*)d_in[4];   // [D]
    float*       out   = (float*)      d_out;     // [B, V]

    float* ws    = (float*)d_ws;
    float* m_all = ws;                                      // 4*B*M*D floats (52.4 MB)
    float* u     = ws + (size_t)4 * NB * NM * ND;           // B*D floats

    // 1) position-encoded story memories for all 4 tables (each used 1-2x across hops)
    embed_story_kernel<<<NB * NM, 128, 0, stream>>>(story, C, m_all);
    // 2) u0 from query
    embed_query_kernel<<<NB, 128, 0, stream>>>(query, C, u);
    // 3) three memory hops (serial dependence through u)
    for (int hop = 0; hop < NHOP; ++hop)
        hop_kernel<<<NB, 256, 0, stream>>>(m_all, u, Tw, Tb, hop);
    // 4) a_hat = u @ C3^T with f32 WMMA  (50000 = 625 * 80, 128 = 8 * 16)
    ahat_wmma_kernel<<<dim3(625, 8), 32, 0, stream>>>(
        u, C + (size_t)3 * NV * ND, out);
}